// RetNetRetention_18846316494924
// MI455X (gfx1250) — compile-verified
//
#include <hip/hip_runtime.h>

// Problem constants (from reference): B=4, S=4096, D=2048, H=16, hd=128
#define B_    4
#define S_    4096
#define D_    2048
#define HD_   128
#define G_    16        // sequence segments
#define LSEG_ 256       // steps per segment (S_/G_)
#define NCH_  16        // 16-step WMMA chunks per segment (LSEG_/16)

typedef __attribute__((ext_vector_type(2))) float v2f;
typedef __attribute__((ext_vector_type(8))) float v8f;

// d^e for per-lane e in [0,16], branchless square-and-multiply (no indexed
// local array -> no scratch spill; values are wave-uniform, index is not).
__device__ __forceinline__ float dpow_u16(float d, int e) {
  float r = 1.f, p = d;
#pragma unroll
  for (int bit = 0; bit < 5; ++bit) {
    r = (e & (1 << bit)) ? r * p : r;
    p = p * p;
  }
  return r;
}

// ---------------------------------------------------------------------------
// Pass 1: per-segment final local state  f[g,b,c] = sum_j d^{LSEG-1-j} k*v
// Pure streaming FMA chain, coalesced across c. 131072 threads = 4096 waves.
// ---------------------------------------------------------------------------
__global__ void __launch_bounds__(256) retnet_seg_finals(
    const float* __restrict__ k, const float* __restrict__ v,
    const float* __restrict__ decay, float* __restrict__ fin) {
  const int tid = blockIdx.x * 256 + threadIdx.x;   // [0, G_*B_*D_)
  const int c  = tid & (D_ - 1);
  const int gb = tid >> 11;                         // / D_
  const int b  = gb & (B_ - 1);
  const int g  = gb >> 2;                           // / B_
  const float d = decay[c >> 7];                    // per-head decay

  const size_t base = (size_t)(b * S_ + g * LSEG_) * D_ + c;
  const float* kp = k + base;
  const float* vp = v + base;

  float f = 0.f;
#pragma unroll 8
  for (int j = 0; j < LSEG_; ++j) {
    f = fmaf(d, f, kp[(size_t)j * D_] * vp[(size_t)j * D_]);
  }
  fin[tid] = f;   // layout [g][b][c]
}

// ---------------------------------------------------------------------------
// Pass 2: WMMA chunked scan. One wave owns (g, b, 16 channels of one head).
// r_chunk = T(d) @ (k*v)  +  d^{t+1} * carry,   out = q .* r_chunk
// T is 16x16 lower-triangular of decay powers -> 4x v_wmma_f32_16x16x4_f32.
// 8192 waves total.
// ---------------------------------------------------------------------------
__global__ void __launch_bounds__(256) retnet_wmma_scan(
    const float* __restrict__ q, const float* __restrict__ k,
    const float* __restrict__ v, const float* __restrict__ decay,
    const float* __restrict__ fin, float* __restrict__ out) {
  const int lane = threadIdx.x & 31;
  const int w    = blockIdx.x * 8 + (threadIdx.x >> 5);  // global wave id
  const int tile = w & 127;                              // D_/16 tiles
  const int gb   = w >> 7;
  const int b    = gb & (B_ - 1);
  const int g    = gb >> 2;
  const int m    = lane & 15;   // row index within 16 (M / N position)
  const int hi   = lane >> 4;   // which half of the wave
  const int c0   = tile << 4;
  const int c    = c0 + m;

  const float d  = decay[c0 >> 7];
  const float d2 = d * d, d4 = d2 * d2, d8 = d4 * d4, d16 = d8 * d8;

  // --- A operand: T[m, j] = (j<=m) ? d^{m-j} : 0, in 16x4 fp32 A layout.
  // WMMA kk covers K-slice j = 4kk..4kk+3; VGPR0 holds K=0 (lanes<16) / K=2
  // (lanes>=16), VGPR1 holds K=1 / K=3  =>  j0 = 4kk + 2*hi, j1 = j0+1.
  v2f amat[4];
#pragma unroll
  for (int kk = 0; kk < 4; ++kk) {
    const int j0 = 4 * kk + 2 * hi;
    const int e0 = m - j0, e1 = m - j0 - 1;
    amat[kk].x = (e0 >= 0) ? dpow_u16(d, e0) : 0.f;
    amat[kk].y = (e1 >= 0) ? dpow_u16(d, e1) : 0.f;
  }

  // --- carry-in for this segment from upstream segment finals (dseg = d^256)
  const float d32 = d16 * d16, d64 = d32 * d32, d128 = d64 * d64;
  const float dseg = d128 * d128;
  float carry = 0.f;
  for (int gp = 0; gp < g; ++gp)
    carry = fmaf(dseg, carry, fin[(size_t)(gp * B_ + b) * D_ + c]);

  // per-lane base decay for C-init: acc[i] scales carry by d^{t+1}, t = i+8*hi
  const float cbase = hi ? (d * d8) : d;   // d^{1+8*hi}

  size_t rowc = (size_t)(b * S_ + g * LSEG_) * D_ + c;
  const float* kb = k + rowc;
  const float* vb = v + rowc;
  const float* qb = q + rowc;
  float*       ob = out + rowc;

  for (int ch = 0; ch < NCH_; ++ch) {
    if (ch + 1 < NCH_) {                       // prefetch next chunk's rows
      __builtin_prefetch(kb + (size_t)16 * D_, 0, 1);
      __builtin_prefetch(vb + (size_t)16 * D_, 0, 1);
      __builtin_prefetch(qb + (size_t)16 * D_, 0, 1);
    }

    // --- B operand: inc[j, c] = k*v in 4x16 fp32 B layout (rows j0, j0+1)
    v2f bmat[4];
#pragma unroll
    for (int kk = 0; kk < 4; ++kk) {
      const size_t j0 = (size_t)(4 * kk + 2 * hi) * D_;
      const size_t j1 = j0 + D_;
      bmat[kk].x = kb[j0] * vb[j0];
      bmat[kk].y = kb[j1] * vb[j1];
    }

    // --- C init with decayed carry:  acc[i] = d^{(i+8*hi)+1} * carry
    v8f acc;
    {
      float fd = cbase;
#pragma unroll
      for (int i = 0; i < 8; ++i) { acc[i] = fd * carry; fd *= d; }
    }

    // --- 16x16 triangular matmul: 4 chained K=4 fp32 WMMAs
#pragma unroll
    for (int kk = 0; kk < 4; ++kk)
      acc = __builtin_amdgcn_wmma_f32_16x16x4_f32(
          false, amat[kk], false, bmat[kk], (short)0, acc, false, false);

    // --- out = q .* r   (C/D layout: VGPR i <-> t = i + 8*hi, N = lane&15)
#pragma unroll
    for (int i = 0; i < 8; ++i) {
      const size_t ro = (size_t)(i + 8 * hi) * D_;
      ob[ro] = qb[ro] * acc[i];
    }

    // --- carry for next chunk = r[t=15, c] = VGPR7 of lanes 16..31
    carry = __int_as_float(__builtin_amdgcn_ds_bpermute(
        (m + 16) << 2, __float_as_int(acc[7])));

    kb += (size_t)16 * D_;
    vb += (size_t)16 * D_;
    qb += (size_t)16 * D_;
    ob += (size_t)16 * D_;
  }
}

extern "C" void kernel_launch(void* const* d_in, const int* in_sizes, int n_in,
                              void* d_out, int out_size, void* d_ws, size_t ws_size,
                              hipStream_t stream) {
  (void)in_sizes; (void)n_in; (void)out_size; (void)ws_size;
  const float* q     = (const float*)d_in[0];
  const float* k     = (const float*)d_in[1];
  const float* v     = (const float*)d_in[2];
  const float* decay = (const float*)d_in[3];
  float* out = (float*)d_out;
  float* fin = (float*)d_ws;   // G_*B_*D_ floats = 512 KB scratch

  // Pass 1: 131072 threads
  retnet_seg_finals<<<dim3((G_ * B_ * D_) / 256), dim3(256), 0, stream>>>(
      k, v, decay, fin);
  // Pass 2: 8192 waves (8 waves / block)
  retnet_wmma_scan<<<dim3((G_ * B_ * (D_ / 16)) / 8), dim3(256), 0, stream>>>(
      q, k, v, decay, fin, out);
}